// FourierBlock_7988639171114
// MI455X (gfx1250) — compile-verified
//
#include <hip/hip_runtime.h>

// FourierBlock for MI455X (gfx1250): three chained GEMMs per (node, head),
// all on V_WMMA_F32_16X16X4_F32 (wave32 WMMA), with the X panel DMA'd into
// LDS by the Tensor Data Mover (TENSOR_LOAD_TO_LDS + s_wait_tensorcnt).
//   stage 1: X(8x16 x 64) * F(64 x 16) -> Re/Im of DFT bins {1,4,5} (cols 0..5)
//   stage 2: [ReX;ImX](16x16) * W1/W2  -> complex channel mixing (per mode)
//   stage 3: Y(128 x 8) * G(8 x 64)    -> irfft with bins 0,1,2 populated
// HBM-bound: ~573 MB moved -> ~25 us floor at 23.3 TB/s; WMMA work is noise.

typedef __attribute__((ext_vector_type(2))) float v2f;
typedef __attribute__((ext_vector_type(8))) float v8f;
typedef __attribute__((ext_vector_type(4))) unsigned int v4u;
typedef __attribute__((ext_vector_type(8))) int v8i;
typedef __attribute__((ext_vector_type(4))) int v4i;

#define NN 2000
#define BB 8
#define LL 64
#define HH 4
#define EE 16
#define PI_F 3.14159265358979f

__device__ __forceinline__ v8f wmma4(v2f a, v2f b, v8f c) {
  // (neg_a, A, neg_b, B, c_mod, C, reuse_a, reuse_b)
  return __builtin_amdgcn_wmma_f32_16x16x4_f32(false, a, false, b, (short)0, c,
                                               false, false);
}

__global__ __launch_bounds__(32) void fourier_block_kernel(
    const float* __restrict__ q, const float* __restrict__ w1,
    const float* __restrict__ w2, float* __restrict__ out) {
  const int n    = blockIdx.x;   // node
  const int h    = blockIdx.y;   // head
  const int lane = threadIdx.x;  // 0..31
  const int hl   = lane >> 4;    // lane half (selects K within A/B operands)
  const int ln   = lane & 15;    // A row / B-D column

  __shared__ __align__(16) float Xs[BB * LL * EE];     // [b][l][i]  32 KB (TDM dest)
  __shared__ float O1[BB * EE * 17];                   // [b][i][col]  8.5 KB
  __shared__ __align__(16) float Ws[2 * EE * EE * 3];  // w1|w2 [i][o][m] 6 KB
  __shared__ float PQ[3 * 2 * 16 * 16];                // [m][P/Q][row][col] 6 KB

  // -------- Tensor Data Mover: DMA the whole X(n,h) panel into LDS ---------
  // 3D tile: dim0 = 16 floats (i, contiguous), dim1 = 64 (l, stride 64 elems),
  // dim2 = 8 (b, stride N*L*64 = 8,192,000 elems). LDS fill order (x,y,z)
  // linear == Xs[b][l][i]. data_size=4B. tensor dims == tile dims (no OOB).
  {
    const unsigned long long ga =
        (unsigned long long)(uintptr_t)q +
        4ull * ((unsigned long long)n * (LL * HH * EE) +
                (unsigned long long)h * EE);
    const unsigned int ldsa = (unsigned int)(uintptr_t)&Xs[0];
    v4u g0;
    g0[0] = 1u;                                   // count=1, user mode
    g0[1] = ldsa;                                 // lds_addr (bytes)
    g0[2] = (unsigned int)(ga & 0xFFFFFFFFu);     // global_addr[31:0]
    g0[3] = (unsigned int)(((ga >> 32) & 0x01FFFFFFu) | 0x80000000u); // +type=2
    v8i g1;
    g1[0] = (int)0x00020000;        // wg_mask=0 | data_size=2 (4B) | no pad
    g1[1] = (int)(16u << 16);       // tensor_dim0[15:0]=16 in [63:48]
    g1[2] = (int)(64u << 16);       // tensor_dim0 hi=0 | tensor_dim1[15:0]=64
    g1[3] = (int)(16u << 16);       // tensor_dim1 hi=0 | tile_dim0=16
    g1[4] = (int)(64u | (8u << 16));// tile_dim1=64 | tile_dim2=8
    g1[5] = (int)64;                // tensor_dim0_stride[31:0] = 64 elems
    g1[6] = 0;                      // stride0 hi | tensor_dim1_stride[15:0]=0
    g1[7] = (int)0x7D;              // tensor_dim1_stride[47:16] (8,192,000>>16)
    v4i g2;
    g2[0] = 8;                      // tensor_dim2 = 8
    g2[1] = 0; g2[2] = 0; g2[3] = 0;
    v4i g3; g3[0] = 0; g3[1] = 0; g3[2] = 0; g3[3] = 0;
#if __clang_major__ >= 23
    v8i g4; g4[0]=0; g4[1]=0; g4[2]=0; g4[3]=0; g4[4]=0; g4[5]=0; g4[6]=0; g4[7]=0;
    __builtin_amdgcn_tensor_load_to_lds(g0, g1, g2, g3, g4, 0);
#else
    __builtin_amdgcn_tensor_load_to_lds(g0, g1, g2, g3, 0);
#endif
  }

  // ------- stage-1 B operand: forward DFT coefficients (overlaps DMA) -------
  // B1[k=l][col]: col 0..2 -> cos(2*pi*mode*l/64), col 3..5 -> -sin(...), else 0
  float b1v[32];
  {
    const int cm    = (ln < 3) ? ln : (ln < 6 ? ln - 3 : 0);
    const int mode  = (cm == 0) ? 1 : (cm == 1 ? 4 : 5);
    const bool useC = (ln < 3);
    const bool useS = (ln >= 3 && ln < 6);
#pragma unroll
    for (int s = 0; s < 16; ++s) {
#pragma unroll
      for (int d = 0; d < 2; ++d) {
        const int l = 4 * s + 2 * hl + d;
        float sn, cs;
        __sincosf(-2.0f * PI_F * (float)(mode * l) / 64.0f, &sn, &cs);
        b1v[2 * s + d] = useC ? cs : (useS ? sn : 0.0f);
      }
    }
  }

  // -------- stage weights -> LDS (coalesced float4, overlaps DMA) -----------
  {
    const float* w1p = w1 + ((size_t)n * HH + h) * (EE * EE * 3);
    const float* w2p = w2 + ((size_t)n * HH + h) * (EE * EE * 3);
    for (int idx = lane; idx < 192; idx += 32) {  // 768 floats each
      ((float4*)Ws)[idx]       = ((const float4*)w1p)[idx];
      ((float4*)Ws)[192 + idx] = ((const float4*)w2p)[idx];
    }
  }

  // Wait for the TDM transfer, then sync LDS view.
  __builtin_amdgcn_s_wait_tensorcnt(0);
  __syncthreads();

  // ---------------- stage 1: per-batch DFT via WMMA --------------------------
  // rows = i (16), K = l (64, 16 k-steps), cols 0..5 = Re/Im of 3 modes
  for (int b = 0; b < BB; ++b) {
    v8f acc = {0.f, 0.f, 0.f, 0.f, 0.f, 0.f, 0.f, 0.f};
#pragma unroll
    for (int s = 0; s < 16; ++s) {
      const int k0 = 4 * s + 2 * hl;
      v2f a, bb;
      a[0] = Xs[b * (LL * EE) + k0 * EE + ln];        // A[row=i][k=l]
      a[1] = Xs[b * (LL * EE) + (k0 + 1) * EE + ln];
      bb[0] = b1v[2 * s];
      bb[1] = b1v[2 * s + 1];
      acc = wmma4(a, bb, acc);
    }
#pragma unroll
    for (int j = 0; j < 8; ++j)  // D: row = j + 8*hl, col = ln
      O1[b * 272 + (j + 8 * hl) * 17 + ln] = acc[j];
  }
  __syncthreads();

  // ---------------- stage 2: complex channel mixing (per mode) ---------------
  // A2 rows 0..7 = ReX(b), rows 8..15 = ImX(b); K = i (16); B = W1/W2 (16x16)
#pragma unroll
  for (int m = 0; m < 3; ++m) {
    v8f P = {0.f, 0.f, 0.f, 0.f, 0.f, 0.f, 0.f, 0.f};
    v8f Q = {0.f, 0.f, 0.f, 0.f, 0.f, 0.f, 0.f, 0.f};
#pragma unroll
    for (int s = 0; s < 4; ++s) {
      const int k0 = 4 * s + 2 * hl;
      const int b  = ln & 7;
      const int cc = (ln < 8) ? m : (m + 3);
      v2f a, bw1, bw2;
      a[0] = O1[b * 272 + k0 * 17 + cc];
      a[1] = O1[b * 272 + (k0 + 1) * 17 + cc];
      bw1[0] = Ws[k0 * 48 + ln * 3 + m];          // W1[k=i][col=o]
      bw1[1] = Ws[(k0 + 1) * 48 + ln * 3 + m];
      bw2[0] = Ws[768 + k0 * 48 + ln * 3 + m];    // W2
      bw2[1] = Ws[768 + (k0 + 1) * 48 + ln * 3 + m];
      P = wmma4(a, bw1, P);
      Q = wmma4(a, bw2, Q);
    }
#pragma unroll
    for (int j = 0; j < 8; ++j) {
      const int row = j + 8 * hl;
      PQ[(2 * m + 0) * 256 + row * 16 + ln] = P[j];
      PQ[(2 * m + 1) * 256 + row * 16 + ln] = Q[j];
    }
  }
  __syncthreads();

  // ---------------- stage 3: inverse transform (bins 0,1,2) ------------------
  // A3 rows: tile rt -> (o = 2*rt + row/8, b = row%8); K: 0=ReY0, 1=ReY1,
  // 2=ImY1, 3=ReY2, 4=ImY2, 5..7=0.  ReY=P[b]-Q[b+8], ImY=Q[b]+P[b+8].
  const float inv64 = 1.0f / 64.0f;
  for (int ct = 0; ct < 4; ++ct) {
    const int l = ct * 16 + ln;
    float s1, c1, s2, c2;
    __sincosf(2.0f * PI_F * (float)l / 64.0f, &s1, &c1);
    __sincosf(4.0f * PI_F * (float)l / 64.0f, &s2, &c2);
    for (int rt = 0; rt < 8; ++rt) {
      v8f acc = {0.f, 0.f, 0.f, 0.f, 0.f, 0.f, 0.f, 0.f};
#pragma unroll
      for (int s = 0; s < 2; ++s) {
        v2f a, bb;
#pragma unroll
        for (int d = 0; d < 2; ++d) {
          const int k = 4 * s + 2 * hl + d;
          const int o = 2 * rt + (ln >> 3);
          const int b = ln & 7;
          float av;
          if (k == 0) {
            av = PQ[0 * 256 + b * 16 + o] - PQ[1 * 256 + (8 + b) * 16 + o];
          } else if (k <= 4) {
            const int m  = (k + 1) >> 1;
            const int im = ((k & 1) == 0) ? 1 : 0;
            const float top = PQ[(2 * m + im) * 256 + b * 16 + o];
            const float bot = PQ[(2 * m + (1 - im)) * 256 + (8 + b) * 16 + o];
            av = im ? (top + bot) : (top - bot);
          } else {
            av = 0.0f;
          }
          float bv;
          if (k == 0)      bv = inv64;
          else if (k == 1) bv = 2.0f * inv64 * c1;
          else if (k == 2) bv = -2.0f * inv64 * s1;
          else if (k == 3) bv = 2.0f * inv64 * c2;
          else if (k == 4) bv = -2.0f * inv64 * s2;
          else             bv = 0.0f;
          a[d]  = av;
          bb[d] = bv;
        }
        acc = wmma4(a, bb, acc);
      }
      // D: vgpr j -> row j (lanes<16) / j+8 (lanes>=16); per-j stores are
      // 16 consecutive floats per half-wave (64B coalesced runs).
      const int o = 2 * rt + hl;
#pragma unroll
      for (int j = 0; j < 8; ++j) {
        const size_t off =
            ((((size_t)j * NN + n) * HH + h) * EE + o) * LL + ct * 16 + ln;
        out[off] = acc[j];
      }
    }
  }
}

extern "C" void kernel_launch(void* const* d_in, const int* in_sizes, int n_in,
                              void* d_out, int out_size, void* d_ws,
                              size_t ws_size, hipStream_t stream) {
  // setup_inputs order: q, k, v, mask, weights1, weights2 (k/v/mask unused)
  const float* q  = (const float*)d_in[0];
  const float* w1 = (const float*)d_in[4];
  const float* w2 = (const float*)d_in[5];
  float* out = (float*)d_out;
  dim3 grid(NN, HH);
  fourier_block_kernel<<<grid, 32, 0, stream>>>(q, w1, w2, out);
}